// MANORenderer_28501402976440
// MI455X (gfx1250) — compile-verified
//
#include <hip/hip_runtime.h>
#include <cstdint>

// Problem constants (from reference)
#define BS   64
#define NV   5778
#define NF   11538
#define TT   9       // T*T barycentric grid points per face
#define IMH  256
#define IMW  512

#define BLK  256                 // threads per block (8 waves, wave32)
#define CHUNK_DW (BLK * 9)       // dwords produced per chunk = 2304
#define CHUNK_B128 (CHUNK_DW/4)  // 576 aligned 16B transfers per chunk

// ---------------------------------------------------------------------------
// Kernel A: perspective projection of vertices.
//   points[b,v] = proj_xy of vertices[b,v] under cam[b]   -> ws (BS, NV, 2) f32
// ---------------------------------------------------------------------------
__global__ __launch_bounds__(BLK)
void proj_kernel(const float* __restrict__ cam,      // (BS, 15)
                 const float* __restrict__ verts,    // (BS, NV, 3)
                 float*       __restrict__ pts)      // (BS, NV, 2)
{
    int idx = blockIdx.x * blockDim.x + threadIdx.x;
    if (idx >= BS * NV) return;
    int b = idx / NV;

    const float* c  = cam + b * 15;
    const float* vv = verts + (size_t)idx * 3;
    // ccm = diag(1,-1,-1): pts = (x, -y, -z)
    float x =  vv[0];
    float y = -vv[1];
    float z = -vv[2];
    // pp = pts . cam_mat^T (row-major 3x3 in c[0..8])
    float pp0 = c[0]*x + c[1]*y + c[2]*z;
    float pp1 = c[3]*x + c[4]*y + c[5]*z;
    float pp2 = c[6]*x + c[7]*y + c[8]*z;
    float X = pp0 / pp2;
    float Y = pp1 / pp2;
    // trans (2x3) in c[9..14], then /255*2 - 1
    float px = c[9]  * X + c[10] * Y + c[11];
    float py = c[12] * X + c[13] * Y + c[14];
    px = px * (2.0f / 255.0f) - 1.0f;
    py = py * (2.0f / 255.0f) - 1.0f;

    float2* o = (float2*)pts;
    o[idx] = make_float2(px, py);
}

// ---------------------------------------------------------------------------
// Kernel B: per-(face, grid-point) bilinear texture sampling.
// - Batch point table (46224 B) staged to LDS via global_load_async_to_lds_b128
// - Vertex gathers served from LDS (ds_load_b64)
// - Output drained via LDS staging + global_store_async_from_lds_b128
//   (576 fully contiguous, 16B-aligned transfers per 256-item chunk)
// ---------------------------------------------------------------------------
__global__ __launch_bounds__(BLK)
void sample_kernel(const float* __restrict__ pts_ws,   // (BS, NV, 2)
                   const float* __restrict__ uv,       // (BS, 3, H, W)
                   const float* __restrict__ coords,   // (2, 9)
                   const int*   __restrict__ faces,    // (BS, NF, 3)
                   float*       __restrict__ out)      // (BS, NF, 3,3,3,3)
{
    __shared__ float lds_pts[NV * 2];      // 46224 B: projected points of batch b
    __shared__ float lds_out[CHUNK_DW];    //  9216 B: output staging for one chunk

    const int b   = blockIdx.y;
    const int tid = threadIdx.x;

    // ---- async stage: global (batch point table) -> LDS, 16B per lane ----
    const float* bpts = pts_ws + (size_t)b * (NV * 2);
    const unsigned lds_pts_base = (unsigned)(unsigned long long)(&lds_pts[0]);
    const unsigned lds_out_base = (unsigned)(unsigned long long)(&lds_out[0]);
    const int n128 = (NV * 2 * 4) / 16;    // 2889 b128 transfers, 16B aligned
    for (int i = tid; i < n128; i += BLK) {
        unsigned     loff = lds_pts_base + (unsigned)i * 16u;
        const float* src  = bpts + i * 4;
        asm volatile("global_load_async_to_lds_b128 %0, %1, off"
                     :: "v"(loff), "v"(src)
                     : "memory");
    }
    asm volatile("s_wait_asynccnt 0" ::: "memory");
    __syncthreads();

    // ---- sampling loop over (face, grid) work items of this batch ----
    const int    total  = NF * TT;                  // 103842
    const int    stride = gridDim.x * BLK;
    const float* img    = uv + (size_t)b * 3 * IMH * IMW;
    const int*   bface  = faces + (size_t)b * NF * 3;
    float*       bout   = out + (size_t)b * NF * 81;

    for (int base = blockIdx.x * BLK; base < total; base += stride) {
        const int  w    = base + tid;
        const bool full = (base + BLK) <= total;    // block-uniform

        float v0 = 0.0f, v1 = 0.0f, v2 = 0.0f;
        if (w < total) {
            int f = w / TT;
            int g = w - f * TT;

            float a  = coords[g];        // coords[0][g]
            float bb = coords[TT + g];   // coords[1][g]

            int i0 = bface[f * 3 + 0];
            int i1 = bface[f * 3 + 1];
            int i2 = bface[f * 3 + 2];

            float p0x = lds_pts[2 * i0], p0y = lds_pts[2 * i0 + 1];
            float p1x = lds_pts[2 * i1], p1y = lds_pts[2 * i1 + 1];
            float p2x = lds_pts[2 * i2], p2y = lds_pts[2 * i2 + 1];

            // sampler = p2 + a*(p0 - p2) + b*(p1 - p2), clipped to [-1,1]
            float sx = p2x + a * (p0x - p2x) + bb * (p1x - p2x);
            float sy = p2y + a * (p0y - p2y) + bb * (p1y - p2y);
            sx = fminf(1.0f, fmaxf(-1.0f, sx));
            sy = fminf(1.0f, fmaxf(-1.0f, sy));

            // grid_sample, bilinear, align_corners=False, zero padding
            float xf = (sx + 1.0f) * (IMW * 0.5f) - 0.5f;
            float yf = (sy + 1.0f) * (IMH * 0.5f) - 0.5f;
            float x0f = floorf(xf), y0f = floorf(yf);
            float wx = xf - x0f,    wy = yf - y0f;
            int x0 = (int)x0f, y0 = (int)y0f;
            int x1 = x0 + 1,   y1 = y0 + 1;

            bool vx0 = (x0 >= 0) && (x0 < IMW);
            bool vx1 = (x1 >= 0) && (x1 < IMW);
            bool vy0 = (y0 >= 0) && (y0 < IMH);
            bool vy1 = (y1 >= 0) && (y1 < IMH);

            int cx0 = min(max(x0, 0), IMW - 1);
            int cx1 = min(max(x1, 0), IMW - 1);
            int cy0 = min(max(y0, 0), IMH - 1);
            int cy1 = min(max(y1, 0), IMH - 1);

            float w00 = (1.0f - wx) * (1.0f - wy) * ((vx0 && vy0) ? 1.0f : 0.0f);
            float w01 = wx          * (1.0f - wy) * ((vx1 && vy0) ? 1.0f : 0.0f);
            float w10 = (1.0f - wx) * wy          * ((vx0 && vy1) ? 1.0f : 0.0f);
            float w11 = wx          * wy          * ((vx1 && vy1) ? 1.0f : 0.0f);

            size_t r0 = (size_t)cy0 * IMW;
            size_t r1 = (size_t)cy1 * IMW;

            const float* c0 = img;
            const float* c1 = img + (size_t)(IMH * IMW);
            const float* c2 = img + (size_t)(2 * IMH * IMW);
            v0 = c0[r0+cx0]*w00 + c0[r0+cx1]*w01 + c0[r1+cx0]*w10 + c0[r1+cx1]*w11;
            v1 = c1[r0+cx0]*w00 + c1[r0+cx1]*w01 + c1[r1+cx0]*w10 + c1[r1+cx1]*w11;
            v2 = c2[r0+cx0]*w00 + c2[r0+cx1]*w01 + c2[r1+cx0]*w10 + c2[r1+cx1]*w11;
        }

        if (full) {
            // stage 9 floats/thread in chunk-contiguous layout (stride 9 ->
            // conflict-free across 32 lanes since gcd(9,64)=1)
            float* lo = &lds_out[tid * 9];
            #pragma unroll
            for (int k = 0; k < 3; ++k) {
                lo[k * 3 + 0] = v0;
                lo[k * 3 + 1] = v1;
                lo[k * 3 + 2] = v2;
            }
            __syncthreads();

            // async drain LDS -> global: 576 contiguous 16B transfers.
            // base is a multiple of 256 -> byte offset base*36 is 16B aligned.
            const float* gdst0 = bout + (size_t)base * 9;
            for (int i = tid; i < CHUNK_B128; i += BLK) {
                unsigned     loff = lds_out_base + (unsigned)i * 16u;
                const float* gp   = gdst0 + i * 4;
                asm volatile("global_store_async_from_lds_b128 %0, %1, off"
                             :: "v"(gp), "v"(loff)
                             : "memory");
            }
            asm volatile("s_wait_asynccnt 0" ::: "memory");
            __syncthreads();   // lds_out reusable next iteration
        } else if (w < total) {
            // ragged tail chunk: direct scalar stores
            float* op = bout + (size_t)w * 9;
            #pragma unroll
            for (int k = 0; k < 3; ++k) {
                op[k * 3 + 0] = v0;
                op[k * 3 + 1] = v1;
                op[k * 3 + 2] = v2;
            }
        }
    }
}

// ---------------------------------------------------------------------------
// Launch
// ---------------------------------------------------------------------------
extern "C" void kernel_launch(void* const* d_in, const int* in_sizes, int n_in,
                              void* d_out, int out_size, void* d_ws, size_t ws_size,
                              hipStream_t stream) {
    const float* cam    = (const float*)d_in[0];   // (BS, 15)
    const float* verts  = (const float*)d_in[1];   // (BS, NV, 3)
    const float* uv     = (const float*)d_in[2];   // (BS, 3, H, W)
    const float* coords = (const float*)d_in[3];   // (2, 9)
    const int*   faces  = (const int*)d_in[4];     // (BS, NF, 3)
    float*       out    = (float*)d_out;           // (BS, NF, 3,3,3,3)
    float*       pts    = (float*)d_ws;            // (BS, NV, 2) scratch, ~2.96 MB

    // Kernel A: project all vertices
    int nA = BS * NV;
    proj_kernel<<<(nA + BLK - 1) / BLK, BLK, 0, stream>>>(cam, verts, pts);

    // Kernel B: 64 blocks per batch x 64 batches, 256 threads each
    dim3 grid(64, BS);
    sample_kernel<<<grid, BLK, 0, stream>>>(pts, uv, coords, faces, out);
}